// P2VModule_26259430048620
// MI455X (gfx1250) — compile-verified
//
#include <hip/hip_runtime.h>

// CDNA5 / gfx1250 wave32 WMMA implementation.
// Pipeline:
//   K0: zero 272-float accumulator region in ws
//   K1: stats   — S = sum(g g^T) (16x16) and sum(g) (16) via v_wmma_f32_16x16x4_f32,
//                 4-way unrolled with independent accumulators for memory-level parallelism
//   K2: bn prep — fold batchnorm into per-feature alpha/shift using S, sum(g), W
//   K3: main    — per-voxel fused gather + WMMA GEMM + BN + ReLU + geo-modulated
//                 masked mean pool + residual add, one wave per voxel.
//
// ws float layout: [0..255]=S, [256..271]=sumc, [272..335]=alpha, [336..399]=shift

typedef __attribute__((ext_vector_type(2))) float v2f;
typedef __attribute__((ext_vector_type(8))) float v8f;

__global__ void zero_ws_kernel(float* __restrict__ ws) {
    int i = blockIdx.x * blockDim.x + threadIdx.x;
    if (i < 272) ws[i] = 0.0f;
}

// ---------------------------------------------------------------------------
// K1: second-moment accumulation  S += G^T G  (WMMA, A-fragment == B-fragment)
// A(16x4) lane L: M = L%16 (channel), VGPR0 = {K0,K2}, VGPR1 = {K1,K3}
// so lane (h = L>>4, c = L&15) loads channel c of samples (4g+2h) and (4g+2h+1).
// 4 groups (16 samples) per iteration into 4 independent accumulators.
// ---------------------------------------------------------------------------
__global__ __launch_bounds__(256) void stats_kernel(
    const float* __restrict__ p_features,  // [N,16]
    const int*   __restrict__ v_map,       // [M*NS]
    float* __restrict__ gacc,              // ws
    int n_groups)                          // (M*NS)/4
{
    __shared__ float sAcc[272];
    for (int i = threadIdx.x; i < 272; i += 256) sAcc[i] = 0.0f;
    __syncthreads();

    const int lane = threadIdx.x & 31;
    const int wave = threadIdx.x >> 5;
    const int h    = lane >> 4;
    const int c    = lane & 15;
    const int wg   = blockIdx.x * 8 + wave;
    const int nw   = gridDim.x * 8;

    v8f S0 = {}, S1 = {}, S2 = {}, S3 = {};
    float sumc = 0.0f;

    const int n_quads = n_groups >> 2;  // 4 groups = 16 samples per iteration
    for (int q = wg; q < n_quads; q += nw) {
        const int base = q * 16 + 2 * h;
        // issue all index loads, then all gathers: 8 index + 8 gather loads per iteration
        const int i00 = v_map[base + 0],  i01 = v_map[base + 1];
        const int i10 = v_map[base + 4],  i11 = v_map[base + 5];
        const int i20 = v_map[base + 8],  i21 = v_map[base + 9];
        const int i30 = v_map[base + 12], i31 = v_map[base + 13];
        v2f a0, a1, a2, a3;
        a0.x = p_features[i00 * 16 + c];  a0.y = p_features[i01 * 16 + c];
        a1.x = p_features[i10 * 16 + c];  a1.y = p_features[i11 * 16 + c];
        a2.x = p_features[i20 * 16 + c];  a2.y = p_features[i21 * 16 + c];
        a3.x = p_features[i30 * 16 + c];  a3.y = p_features[i31 * 16 + c];
        sumc += ((a0.x + a0.y) + (a1.x + a1.y)) + ((a2.x + a2.y) + (a3.x + a3.y));
        S0 = __builtin_amdgcn_wmma_f32_16x16x4_f32(false, a0, false, a0, (short)0, S0, false, false);
        S1 = __builtin_amdgcn_wmma_f32_16x16x4_f32(false, a1, false, a1, (short)0, S1, false, false);
        S2 = __builtin_amdgcn_wmma_f32_16x16x4_f32(false, a2, false, a2, (short)0, S2, false, false);
        S3 = __builtin_amdgcn_wmma_f32_16x16x4_f32(false, a3, false, a3, (short)0, S3, false, false);
    }
    // tail: remaining 0..3 groups, one per wave
    for (int g = (n_quads << 2) + wg; g < n_groups; g += nw) {
        const int base = g * 4 + 2 * h;
        const int i0 = v_map[base];
        const int i1 = v_map[base + 1];
        v2f a;
        a.x = p_features[i0 * 16 + c];
        a.y = p_features[i1 * 16 + c];
        sumc += a.x + a.y;
        S0 = __builtin_amdgcn_wmma_f32_16x16x4_f32(false, a, false, a, (short)0, S0, false, false);
    }

    // combine accumulators; D layout: lane L, VGPR j -> S[j + 8*(L>>4)][L&15]
    #pragma unroll
    for (int j = 0; j < 8; ++j) {
        const float s = (S0[j] + S1[j]) + (S2[j] + S3[j]);
        atomicAdd(&sAcc[(j + 8 * h) * 16 + c], s);
    }
    atomicAdd(&sAcc[256 + c], sumc);
    __syncthreads();

    for (int i = threadIdx.x; i < 272; i += 256)
        atomicAdd(&gacc[i], sAcc[i]);
}

// ---------------------------------------------------------------------------
// K2: fold batchnorm into alpha (scale) / shift:  y = alpha*x + shift
//   mu_f  = W_f . sumc / NT
//   Ex2_f = W_f^T S W_f / NT ;  var = Ex2 - mu^2
// ---------------------------------------------------------------------------
__global__ void bn_param_kernel(
    const float* __restrict__ W,      // [64,16]
    const float* __restrict__ gamma,  // [64]
    const float* __restrict__ beta,   // [64]
    float* __restrict__ ws,
    float inv_nt)
{
    int f = threadIdx.x;
    if (f >= 64) return;
    const float* S    = ws;
    const float* sumc = ws + 256;

    float wrow[16];
    #pragma unroll
    for (int c = 0; c < 16; ++c) wrow[c] = W[f * 16 + c];

    float mu = 0.0f;
    #pragma unroll
    for (int c = 0; c < 16; ++c) mu += wrow[c] * sumc[c];
    mu *= inv_nt;

    float ex2 = 0.0f;
    #pragma unroll
    for (int i = 0; i < 16; ++i) {
        float acc = 0.0f;
        #pragma unroll
        for (int j = 0; j < 16; ++j) acc += wrow[j] * S[i * 16 + j];
        ex2 += wrow[i] * acc;
    }
    ex2 *= inv_nt;

    const float var   = ex2 - mu * mu;
    const float scale = gamma[f] * rsqrtf(var + 1e-5f);
    ws[272 + f] = scale;
    ws[336 + f] = beta[f] - mu * scale;
}

// ---------------------------------------------------------------------------
// K3: fused main kernel. One wave32 per voxel, 8 voxels per 256-thread block.
// Per voxel: 2 sample-groups x 4 feature-tiles x 4 k-steps = 32 WMMA f32 ops.
// ---------------------------------------------------------------------------
__global__ __launch_bounds__(256) void p2v_main_kernel(
    const float* __restrict__ p_coords,    // [N,3]
    const float* __restrict__ p_features,  // [N,16]
    const int*   __restrict__ v_indices,   // [M,4]
    const float* __restrict__ v_features,  // [M,64]
    const int*   __restrict__ v_map,       // [M,32]
    const int*   __restrict__ v_mask,      // [M,32]
    const float* __restrict__ W,           // [64,16]
    const float* __restrict__ ws,          // alpha @272, shift @336
    float* __restrict__ out,               // [M,64]
    int M)
{
    __shared__ float sGeo[8 * 128];  // per-wave: 32 samples x 4 geo comps (mask folded in)

    const int lane = threadIdx.x & 31;
    const int wave = threadIdx.x >> 5;
    const int m    = blockIdx.x * 8 + wave;
    const bool active = (m < M);
    const int h = lane >> 4;   // half-wave
    const int r = lane & 15;   // row/col within fragment

    // Stage W as B-matrix fragments: B[ft][kk], lane L: col = r (feature),
    // values W[16*ft + r][4*kk + 2*h (+1)]  (voxel-invariant).
    v2f B[4][4];
    #pragma unroll
    for (int ft = 0; ft < 4; ++ft)
        #pragma unroll
        for (int kk = 0; kk < 4; ++kk) {
            const float* src = W + (ft * 16 + r) * 16 + kk * 4 + 2 * h;
            B[ft][kk].x = src[0];
            B[ft][kk].y = src[1];
        }

    float alpha4[4], shift4[4];
    #pragma unroll
    for (int ft = 0; ft < 4; ++ft) {
        alpha4[ft] = ws[272 + ft * 16 + r];
        shift4[ft] = ws[336 + ft * 16 + r];
    }

    // hoist residual loads (lanes h==0 are the writers)
    float vf[4] = {0.0f, 0.0f, 0.0f, 0.0f};
    if (active && h == 0) {
        #pragma unroll
        for (int ft = 0; ft < 4; ++ft)
            vf[ft] = v_features[m * 64 + ft * 16 + r];
    }

    float den = 1.0f;
    int pidx_all = 0;  // lane L holds v_map[m*32 + L]
    if (active) {
        // voxel center (V_SIZE = {0.1,0.1,0.15}, PC_MIN = {0,-40,-3})
        const int vi1 = v_indices[m * 4 + 1];
        const int vi2 = v_indices[m * 4 + 2];
        const int vi3 = v_indices[m * 4 + 3];
        const float cx = ((float)vi3 + 0.5f) * 0.10f + 0.0f;
        const float cy = ((float)vi2 + 0.5f) * 0.10f - 40.0f;
        const float cz = ((float)vi1 + 0.5f) * 0.15f - 3.0f;

        // lane L handles sample s = L: geo kernel, mask folded in
        const float mk = (float)v_mask[m * 32 + lane];
        pidx_all = v_map[m * 32 + lane];
        const float px = p_coords[pidx_all * 3 + 0];
        const float py = p_coords[pidx_all * 3 + 1];
        const float pz = p_coords[pidx_all * 3 + 2];
        const float rx = px - cx, ry = py - cy, rz = pz - cz;
        const float d2 = rx * rx + ry * ry + rz * rz;
        float4 gm;
        gm.x = rx * (1.0f / 0.10f)   * mk;
        gm.y = ry * (1.0f / 0.10f)   * mk;
        gm.z = rz * (1.0f / 0.15f)   * mk;
        gm.w = d2 * (1.0f / 0.0425f) * mk;   // sum(V_SIZE^2) = 0.0425
        *(float4*)&sGeo[wave * 128 + lane * 4] = gm;

        // den = clip(sum(mask), 1)
        float dd = mk;
        dd += __shfl_xor(dd, 16);
        dd += __shfl_xor(dd, 8);
        dd += __shfl_xor(dd, 4);
        dd += __shfl_xor(dd, 2);
        dd += __shfl_xor(dd, 1);
        den = fmaxf(dd, 1.0f);
    }
    __syncthreads();

    if (active) {
        const int   kcomp = lane & 3;   // == (16*ft + r) % 4 for every ft
        float numacc[4] = {0.0f, 0.0f, 0.0f, 0.0f};

        #pragma unroll
        for (int t = 0; t < 2; ++t) {
            // A-fragment gather: lane L -> row r (sample t*16+r), K pair {2h, 2h+1};
            // the point index is already resident in the wave -> cross-lane fetch.
            const int pidx = __shfl(pidx_all, t * 16 + r);
            const float* prow = p_features + pidx * 16 + 2 * h;
            v2f A[4];
            #pragma unroll
            for (int kk = 0; kk < 4; ++kk) {
                A[kk].x = prow[kk * 4 + 0];
                A[kk].y = prow[kk * 4 + 1];
            }
            // per-lane geo values for its 8 D-fragment slots
            float g8[8];
            #pragma unroll
            for (int j = 0; j < 8; ++j)
                g8[j] = sGeo[wave * 128 + (t * 16 + 8 * h + j) * 4 + kcomp];

            #pragma unroll
            for (int ft = 0; ft < 4; ++ft) {
                v8f D = {};
                D = __builtin_amdgcn_wmma_f32_16x16x4_f32(false, A[0], false, B[ft][0], (short)0, D, false, false);
                D = __builtin_amdgcn_wmma_f32_16x16x4_f32(false, A[1], false, B[ft][1], (short)0, D, false, false);
                D = __builtin_amdgcn_wmma_f32_16x16x4_f32(false, A[2], false, B[ft][2], (short)0, D, false, false);
                D = __builtin_amdgcn_wmma_f32_16x16x4_f32(false, A[3], false, B[ft][3], (short)0, D, false, false);
                #pragma unroll
                for (int j = 0; j < 8; ++j) {
                    const float y = fmaxf(alpha4[ft] * D[j] + shift4[ft], 0.0f);
                    numacc[ft] += y * g8[j];
                }
            }
        }

        const float invden = 1.0f / den;
        #pragma unroll
        for (int ft = 0; ft < 4; ++ft) {
            const float tot = numacc[ft] + __shfl_xor(numacc[ft], 16);
            if (h == 0) {
                out[m * 64 + ft * 16 + r] = vf[ft] + tot * invden;
            }
        }
    }
}

// ---------------------------------------------------------------------------
extern "C" void kernel_launch(void* const* d_in, const int* in_sizes, int n_in,
                              void* d_out, int out_size, void* d_ws, size_t ws_size,
                              hipStream_t stream) {
    const float* p_coords   = (const float*)d_in[0];
    const float* p_features = (const float*)d_in[1];
    const int*   v_indices  = (const int*)d_in[2];
    const float* v_features = (const float*)d_in[3];
    const int*   v_map      = (const int*)d_in[4];
    const int*   v_mask     = (const int*)d_in[5];
    const float* W          = (const float*)d_in[6];
    const float* gamma      = (const float*)d_in[7];
    const float* beta       = (const float*)d_in[8];
    float* out = (float*)d_out;
    float* ws  = (float*)d_ws;

    const int M        = in_sizes[2] / 4;       // v_indices is [M,4]
    const int MT       = in_sizes[4];           // v_map is [M,NS] -> M*NS samples
    const int n_groups = MT / 4;                // 4 samples per WMMA in stats pass
    const float inv_nt = 1.0f / (float)MT;

    hipLaunchKernelGGL(zero_ws_kernel, dim3(2), dim3(256), 0, stream, ws);
    hipLaunchKernelGGL(stats_kernel, dim3(1024), dim3(256), 0, stream,
                       p_features, v_map, ws, n_groups);
    hipLaunchKernelGGL(bn_param_kernel, dim3(1), dim3(64), 0, stream,
                       W, gamma, beta, ws, inv_nt);
    hipLaunchKernelGGL(p2v_main_kernel, dim3((M + 7) / 8), dim3(256), 0, stream,
                       p_coords, p_features, v_indices, v_features,
                       v_map, v_mask, W, ws, out, M);
}